// GCNWithGIB_85804856640340
// MI455X (gfx1250) — compile-verified
//
#include <hip/hip_runtime.h>
#include <hip/hip_bf16.h>
#include <math.h>

typedef __attribute__((ext_vector_type(2))) float v2f;
typedef __attribute__((ext_vector_type(8))) float v8f;

#define GCN_N 100000
#define GCN_E 3200000
#define GCN_FIN 128
#define GCN_HID 16
#define GCN_C 64
#define GCN_NT (GCN_N / 16)   // 6250 row tiles of 16

// ---------------------------------------------------------------------------
// Counter-based uniform hash -> Gumbel noise (deterministic per edge).
// sigmoid(x) > 0.5  <=>  x > 0, so edge kept iff edge_probs[e] + gumbel > 0.
// ---------------------------------------------------------------------------
__device__ __forceinline__ float uniform_hash(unsigned int e) {
    unsigned int x = e * 0x9E3779B9u + 0x7F4A7C15u;
    x ^= x >> 16; x *= 0x85EBCA6Bu;
    x ^= x >> 13; x *= 0xC2B2AE35u;
    x ^= x >> 16;
    float u = (float)(x >> 8) * (1.0f / 16777216.0f);
    return fminf(fmaxf(u, 1e-7f), 1.0f - 1e-7f);
}

// ---------------------------------------------------------------------------
// 0) zero accumulators, deg = 1 (self loop). Re-run every call.
// ---------------------------------------------------------------------------
__global__ void init_kernel(float* __restrict__ deg, float* __restrict__ agg1,
                            float* __restrict__ agg2) {
    unsigned int i = blockIdx.x * blockDim.x + threadIdx.x;
    if (i < (unsigned)GCN_N * GCN_C) agg2[i] = 0.0f;
    if (i < (unsigned)GCN_N * GCN_HID) agg1[i] = 0.0f;
    if (i < (unsigned)GCN_N) deg[i] = 1.0f;
}

// ---------------------------------------------------------------------------
// 1) edge selection + degree accumulation
// ---------------------------------------------------------------------------
__global__ void edge_mask_kernel(const float* __restrict__ eprob,
                                 const int* __restrict__ dst,
                                 float* __restrict__ ew, float* __restrict__ deg) {
    unsigned int e = blockIdx.x * blockDim.x + threadIdx.x;
    if (e >= GCN_E) return;
    float u = uniform_hash(e);
    float g = -logf(-logf(u));
    bool sel = (eprob[e] + g) > 0.0f;
    ew[e] = sel ? 1.0f : 0.0f;
    if (sel) atomicAdd(&deg[dst[e]], 1.0f);
}

// ---------------------------------------------------------------------------
// 2) deg -> d^{-1/2} in place
// ---------------------------------------------------------------------------
__global__ void dinv_kernel(float* __restrict__ deg) {
    unsigned int n = blockIdx.x * blockDim.x + threadIdx.x;
    if (n < (unsigned)GCN_N) deg[n] = rsqrtf(deg[n]);
}

// ---------------------------------------------------------------------------
// 3) coef[e] = ew * dinv[src]*dinv[dst], folded in place (reused both layers)
// ---------------------------------------------------------------------------
__global__ void coef_kernel(float* __restrict__ coef, const int* __restrict__ src,
                            const int* __restrict__ dst, const float* __restrict__ dinv) {
    unsigned int e = blockIdx.x * blockDim.x + threadIdx.x;
    if (e >= GCN_E) return;
    float w = coef[e];
    coef[e] = (w != 0.0f) ? dinv[src[e]] * dinv[dst[e]] : 0.0f;
}

// ---------------------------------------------------------------------------
// GEMM1: h1[N,16] = x[N,128] @ W1[128,16]  — f32 WMMA 16x16x4, wave/16-row tile
// A 16x4: lane m=lane&15 holds row M=m; kh=lane>>4 selects K pair {2kh,2kh+1}.
// D 16x16: vgpr r holds M = r + 8*kh, N = m.
// ---------------------------------------------------------------------------
__global__ void gemm1_kernel(const float* __restrict__ x, const float* __restrict__ W1,
                             float* __restrict__ h1) {
    __shared__ float sW[GCN_FIN * GCN_HID];   // 8 KB
    for (int i = threadIdx.x; i < GCN_FIN * GCN_HID; i += blockDim.x) sW[i] = W1[i];
    __syncthreads();

    int wave = threadIdx.x >> 5;
    int lane = threadIdx.x & 31;
    int tile = blockIdx.x * 8 + wave;
    if (tile >= GCN_NT) return;                     // wave-uniform exit
    int row0 = tile * 16;
    int m  = lane & 15;
    int kh = lane >> 4;
    const float* xrow = x + (size_t)(row0 + m) * GCN_FIN;

    v8f c = {};
#if __has_builtin(__builtin_amdgcn_wmma_f32_16x16x4_f32)
    for (int k0 = 0; k0 < GCN_FIN; k0 += 4) {
        int ka = k0 + 2 * kh;
        v2f a, b;
        a.x = xrow[ka];             a.y = xrow[ka + 1];
        b.x = sW[ka * GCN_HID + m]; b.y = sW[(ka + 1) * GCN_HID + m];
        c = __builtin_amdgcn_wmma_f32_16x16x4_f32(false, a, false, b,
                                                  (short)0, c, false, false);
    }
#else
    for (int r = 0; r < 8; ++r) {
        float acc = 0.0f;
        const float* xr = x + (size_t)(row0 + r + 8 * kh) * GCN_FIN;
        for (int k = 0; k < GCN_FIN; ++k) acc += xr[k] * sW[k * GCN_HID + m];
        c[r] = acc;
    }
#endif
    float* hp = h1 + (size_t)row0 * GCN_HID;
#pragma unroll
    for (int r = 0; r < 8; ++r)
        hp[(r + 8 * kh) * GCN_HID + m] = c[r];
}

// ---------------------------------------------------------------------------
// 5) scatter layer 1: 16 threads/edge, atomics skipped for pruned edges
// ---------------------------------------------------------------------------
__global__ void scatter1_kernel(const float* __restrict__ h1, const float* __restrict__ coef,
                                const int* __restrict__ src, const int* __restrict__ dst,
                                float* __restrict__ agg1) {
    unsigned int idx = blockIdx.x * blockDim.x + threadIdx.x;
    if (idx >= (unsigned)GCN_E * 16u) return;
    unsigned int e = idx >> 4;
    int f = idx & 15;
    float cf = coef[e];
    if (cf != 0.0f) {
        float v = h1[(size_t)src[e] * GCN_HID + f] * cf;
        atomicAdd(&agg1[(size_t)dst[e] * GCN_HID + f], v);
    }
}

// ---------------------------------------------------------------------------
// 6) node update layer 1: out = relu(agg + h*dinv^2 + b1), in place over agg1
// ---------------------------------------------------------------------------
__global__ void node1_kernel(float* __restrict__ agg1, const float* __restrict__ h1,
                             const float* __restrict__ dinv, const float* __restrict__ b1) {
    unsigned int idx = blockIdx.x * blockDim.x + threadIdx.x;
    if (idx >= (unsigned)GCN_N * GCN_HID) return;
    unsigned int n = idx >> 4;
    int f = idx & 15;
    float di = dinv[n];
    float v = agg1[idx] + h1[idx] * di * di + b1[f];
    agg1[idx] = fmaxf(v, 0.0f);
}

// ---------------------------------------------------------------------------
// GEMM2: h2[N,64] = out1[N,16] @ W2[16,64] — 16 WMMAs per 16x64 tile
// ---------------------------------------------------------------------------
__global__ void gemm2_kernel(const float* __restrict__ h, const float* __restrict__ W2,
                             float* __restrict__ h2) {
    __shared__ float sW[GCN_HID * GCN_C];    // 4 KB
    for (int i = threadIdx.x; i < GCN_HID * GCN_C; i += blockDim.x) sW[i] = W2[i];
    __syncthreads();

    int wave = threadIdx.x >> 5;
    int lane = threadIdx.x & 31;
    int tile = blockIdx.x * 8 + wave;
    if (tile >= GCN_NT) return;
    int row0 = tile * 16;
    int m  = lane & 15;
    int kh = lane >> 4;
    const float* hrow = h + (size_t)(row0 + m) * GCN_HID;

    v8f c[4] = {};
#if __has_builtin(__builtin_amdgcn_wmma_f32_16x16x4_f32)
    for (int k0 = 0; k0 < GCN_HID; k0 += 4) {
        int ka = k0 + 2 * kh;
        v2f a;
        a.x = hrow[ka]; a.y = hrow[ka + 1];
#pragma unroll
        for (int nt = 0; nt < 4; ++nt) {
            v2f b;
            b.x = sW[ka * GCN_C + nt * 16 + m];
            b.y = sW[(ka + 1) * GCN_C + nt * 16 + m];
            c[nt] = __builtin_amdgcn_wmma_f32_16x16x4_f32(false, a, false, b,
                                                          (short)0, c[nt], false, false);
        }
    }
#else
    for (int nt = 0; nt < 4; ++nt)
        for (int r = 0; r < 8; ++r) {
            float acc = 0.0f;
            const float* hr = h + (size_t)(row0 + r + 8 * kh) * GCN_HID;
            for (int k = 0; k < GCN_HID; ++k) acc += hr[k] * sW[k * GCN_C + nt * 16 + m];
            c[nt][r] = acc;
        }
#endif
    float* op = h2 + (size_t)row0 * GCN_C;
#pragma unroll
    for (int nt = 0; nt < 4; ++nt)
#pragma unroll
        for (int r = 0; r < 8; ++r)
            op[(r + 8 * kh) * GCN_C + nt * 16 + m] = c[nt][r];
}

// ---------------------------------------------------------------------------
// 8) scatter layer 2: 32 threads/edge, float2 per thread
// ---------------------------------------------------------------------------
__global__ void scatter2_kernel(const float* __restrict__ h2, const float* __restrict__ coef,
                                const int* __restrict__ src, const int* __restrict__ dst,
                                float* __restrict__ agg2) {
    unsigned int idx = blockIdx.x * blockDim.x + threadIdx.x;
    if (idx >= (unsigned)GCN_E * 32u) return;
    unsigned int e = idx >> 5;
    int p = idx & 31;
    float cf = coef[e];
    if (cf != 0.0f) {
        const float2 hv = *(const float2*)&h2[(size_t)src[e] * GCN_C + 2 * p];
        float* ap = &agg2[(size_t)dst[e] * GCN_C + 2 * p];
        atomicAdd(ap,     hv.x * cf);
        atomicAdd(ap + 1, hv.y * cf);
    }
}

// ---------------------------------------------------------------------------
// 9) node update layer 2 + log_softmax (wave per node, wave32 shuffles)
// ---------------------------------------------------------------------------
__global__ void node2_kernel(const float* __restrict__ agg2, const float* __restrict__ h2,
                             const float* __restrict__ dinv, const float* __restrict__ b2,
                             float* __restrict__ out) {
    int wave = threadIdx.x >> 5;
    int lane = threadIdx.x & 31;
    int n = blockIdx.x * 8 + wave;
    if (n >= GCN_N) return;
    float di = dinv[n];
    float d2 = di * di;
    size_t base = (size_t)n * GCN_C;
    float v0 = agg2[base + lane]      + h2[base + lane]      * d2 + b2[lane];
    float v1 = agg2[base + 32 + lane] + h2[base + 32 + lane] * d2 + b2[32 + lane];

    float mx = fmaxf(v0, v1);
#pragma unroll
    for (int o = 16; o >= 1; o >>= 1) mx = fmaxf(mx, __shfl_xor(mx, o, 32));
    float s = expf(v0 - mx) + expf(v1 - mx);
#pragma unroll
    for (int o = 16; o >= 1; o >>= 1) s += __shfl_xor(s, o, 32);
    float lse = logf(s);

    out[base + lane]      = v0 - mx - lse;
    out[base + 32 + lane] = v1 - mx - lse;
}

// ---------------------------------------------------------------------------
extern "C" void kernel_launch(void* const* d_in, const int* in_sizes, int n_in,
                              void* d_out, int out_size, void* d_ws, size_t ws_size,
                              hipStream_t stream) {
    const float* x     = (const float*)d_in[0];
    const int*   eidx  = (const int*)d_in[1];
    const float* W1    = (const float*)d_in[2];
    const float* b1    = (const float*)d_in[3];
    const float* W2    = (const float*)d_in[4];
    const float* b2    = (const float*)d_in[5];
    const float* eprob = (const float*)d_in[6];
    float*       out   = (float*)d_out;

    const int* src = eidx;            // edge_index[0]
    const int* dst = eidx + GCN_E;    // edge_index[1]

    float* ws   = (float*)d_ws;
    float* coef = ws;                                 // [E]   ew -> coef (in place)
    float* deg  = coef + (size_t)GCN_E;               // [N]   deg -> dinv (in place)
    float* h1   = deg  + (size_t)GCN_N;               // [N,16]
    float* agg1 = h1   + (size_t)GCN_N * GCN_HID;     // [N,16] agg -> relu out (in place)
    float* h2   = agg1 + (size_t)GCN_N * GCN_HID;     // [N,64]
    float* agg2 = h2   + (size_t)GCN_N * GCN_C;       // [N,64]

    const int B = 256;
    init_kernel<<<(GCN_N * GCN_C + B - 1) / B, B, 0, stream>>>(deg, agg1, agg2);
    edge_mask_kernel<<<(GCN_E + B - 1) / B, B, 0, stream>>>(eprob, dst, coef, deg);
    dinv_kernel<<<(GCN_N + B - 1) / B, B, 0, stream>>>(deg);
    coef_kernel<<<(GCN_E + B - 1) / B, B, 0, stream>>>(coef, src, dst, deg);

    gemm1_kernel<<<(GCN_NT + 7) / 8, B, 0, stream>>>(x, W1, h1);
    scatter1_kernel<<<((size_t)GCN_E * 16 + B - 1) / B, B, 0, stream>>>(h1, coef, src, dst, agg1);
    node1_kernel<<<(GCN_N * GCN_HID + B - 1) / B, B, 0, stream>>>(agg1, h1, deg, b1);

    gemm2_kernel<<<(GCN_NT + 7) / 8, B, 0, stream>>>(agg1, W2, h2);
    scatter2_kernel<<<((size_t)GCN_E * 32 + B - 1) / B, B, 0, stream>>>(h2, coef, src, dst, agg2);
    node2_kernel<<<(GCN_N + 7) / 8, B, 0, stream>>>(agg2, h2, deg, b2, out);
}